// MLPBlock_6425271075385
// MI455X (gfx1250) — compile-verified
//
#include <hip/hip_runtime.h>
#include <hip/hip_bf16.h>
#include <math.h>

// Problem dims (fixed by the reference)
#define T_DIM 1024
#define H_DIM 2880
#define I_DIM 2880
#define E_DIM 8
#define TWOI_DIM 5760
#define ALPHA 1.702f
#define LIMIT 7.0f
#define EPS_RMS 1e-5f

typedef float v2f __attribute__((ext_vector_type(2)));
typedef float v8f __attribute__((ext_vector_type(8)));

// ---------------------------------------------------------------------------
// gfx1250 async global->LDS copy (ASYNCcnt path) via inline asm.
// The clang builtin exists but its pointer params are in the unspellable
// LangAS::cuda_device address space, so we emit the instruction directly.
//   global_load_async_to_lds_b128 vDST(lds byte offset), v[addr64], off
// Generic pointers to __shared__ are {aperture_hi, lds_offset}, so the low
// 32 bits are exactly the LDS byte offset the instruction consumes.
// ---------------------------------------------------------------------------
__device__ __forceinline__ void async_copy_b128(float* lds_dst,
                                                const float* glb_src) {
    unsigned lds_off = (unsigned)(unsigned long long)lds_dst;
    asm volatile("global_load_async_to_lds_b128 %0, %1, off"
                 :
                 : "v"(lds_off), "v"(glb_src)
                 : "memory");
}

__device__ __forceinline__ void async_wait_all() {
    asm volatile("s_wait_asynccnt 0x0" ::: "memory");
}

// ---------------------------------------------------------------------------
// Deterministic block-wide sum (256 threads = 8 waves of 32): shfl tree +
// fixed-order LDS combine. No float atomics -> bit-identical across replays.
// ---------------------------------------------------------------------------
__device__ __forceinline__ float block_reduce_sum(float v, float* sred) {
    #pragma unroll
    for (int o = 16; o > 0; o >>= 1) v += __shfl_down(v, o, 32);
    const int w = threadIdx.x >> 5;
    if ((threadIdx.x & 31) == 0) sred[w] = v;
    __syncthreads();
    float r = 0.0f;
    if (threadIdx.x < 8) r = sred[threadIdx.x];
    if (w == 0) {
        #pragma unroll
        for (int o = 4; o > 0; o >>= 1) r += __shfl_down(r, o, 32);
    }
    if (threadIdx.x == 0) sred[0] = r;
    __syncthreads();
    float out = sred[0];
    __syncthreads();
    return out;
}

__device__ __forceinline__ float sigmoid_f(float z) {
    return 1.0f / (1.0f + __expf(-z));
}

// ---------------------------------------------------------------------------
// Kernel 1: RMSNorm + gate logits + top-2 softmax -> t[T,H], combine[T,E]
// One block (256 threads) per token.
// ---------------------------------------------------------------------------
__global__ void gate_rmsnorm_kernel(const float* __restrict__ x,
                                    const float* __restrict__ scale,
                                    const float* __restrict__ gate_w,
                                    const float* __restrict__ gate_b,
                                    float* __restrict__ t,
                                    float* __restrict__ combine) {
    __shared__ float sred[8];
    const int tok = blockIdx.x;
    const float* xr = x + (size_t)tok * H_DIM;
    float* tr = t + (size_t)tok * H_DIM;

    // sum of squares
    float ss = 0.0f;
    for (int j = threadIdx.x; j < H_DIM; j += 256) {
        float v = xr[j];
        ss += v * v;
    }
    ss = block_reduce_sum(ss, sred);
    const float rms = rsqrtf(ss * (1.0f / (float)H_DIM) + EPS_RMS);

    // normalized row + gate dot partials
    float g[E_DIM];
    #pragma unroll
    for (int e = 0; e < E_DIM; ++e) g[e] = 0.0f;
    for (int j = threadIdx.x; j < H_DIM; j += 256) {
        float tv = xr[j] * rms * scale[j];
        tr[j] = tv;
        #pragma unroll
        for (int e = 0; e < E_DIM; ++e) g[e] += tv * gate_w[e * H_DIM + j];
    }
    #pragma unroll
    for (int e = 0; e < E_DIM; ++e) g[e] = block_reduce_sum(g[e], sred);

    if (threadIdx.x == 0) {
        float logits[E_DIM];
        #pragma unroll
        for (int e = 0; e < E_DIM; ++e) logits[e] = g[e] + gate_b[e];
        // top-2 (first occurrence wins ties, like lax.top_k)
        int i1 = 0;
        #pragma unroll
        for (int e = 1; e < E_DIM; ++e) if (logits[e] > logits[i1]) i1 = e;
        int i2 = (i1 == 0) ? 1 : 0;
        #pragma unroll
        for (int e = 0; e < E_DIM; ++e)
            if (e != i1 && logits[e] > logits[i2]) i2 = e;
        float p1 = 1.0f / (1.0f + __expf(logits[i2] - logits[i1]));
        float p2 = 1.0f - p1;
        #pragma unroll
        for (int e = 0; e < E_DIM; ++e) {
            float c = (e == i1) ? p1 : ((e == i2) ? p2 : 0.0f);
            combine[(size_t)tok * E_DIM + e] = c;
        }
    }
}

// ---------------------------------------------------------------------------
// Issue one K-stage (32 wide) of async global->LDS copies.
//   A tile: 128 rows x 32 cols = 1024 b128 / 256 threads (4 each)
//   B tile:  64 rows x 32 cols =  512 b128 / 256 threads (2 each)
// LDS row stride 36 floats: keeps 16B alignment for b128 and gives
// conflict-free transposed reads (gcd(36,64)=4, 9 coprime 16 -> 32 banks).
// ---------------------------------------------------------------------------
__device__ __forceinline__ void stage_issue(const float* __restrict__ Arows,
                                            const float* __restrict__ Brows,
                                            int Kdim, int k0,
                                            float (*As)[36],
                                            float (*Bs)[36]) {
    const int tid = threadIdx.x;
    #pragma unroll
    for (int r = 0; r < 4; ++r) {
        int idx = tid + r * 256;
        int m = idx >> 3;
        int kq = idx & 7;
        async_copy_b128(&As[m][kq * 4],
                        Arows + (size_t)m * Kdim + k0 + kq * 4);
    }
    #pragma unroll
    for (int r = 0; r < 2; ++r) {
        int idx = tid + r * 256;
        int n = idx >> 3;
        int kq = idx & 7;
        async_copy_b128(&Bs[n][kq * 4],
                        Brows + (size_t)n * Kdim + k0 + kq * 4);
    }
}

// ---------------------------------------------------------------------------
// WMMA main loop: C[128x64] += A[128xK] * B[64xK]^T, double-buffered LDS.
// Block = 256 threads = 8 waves; wave (wm in 0..3, wn in 0..1) owns a 32x32
// tile = 2x2 v_wmma_f32_16x16x4_f32 accumulators.
// Pipeline per stage s: wait own ASYNCcnt -> barrier (all waves' LDS data
// resident, and everyone is done computing stage s-1) -> issue stage s+1 into
// the other buffer -> compute stage s. Copies overlap the WMMA stream.
// ---------------------------------------------------------------------------
__device__ __forceinline__ void mma_main_loop(const float* __restrict__ Arows,
                                              const float* __restrict__ Brows,
                                              int Kdim,
                                              float (*As)[128][36],
                                              float (*Bs)[64][36],
                                              v8f acc[2][2]) {
    const int tid = threadIdx.x;
    const int lane = tid & 31;
    const int wave = tid >> 5;
    const int wm = wave & 3;
    const int wn = wave >> 2;
    const int lmod = lane & 15;
    const int lhalf = lane >> 4;

    #pragma unroll
    for (int mi = 0; mi < 2; ++mi)
        #pragma unroll
        for (int ni = 0; ni < 2; ++ni)
            acc[mi][ni] = (v8f){0.f, 0.f, 0.f, 0.f, 0.f, 0.f, 0.f, 0.f};

    const int nstages = Kdim / 32;
    stage_issue(Arows, Brows, Kdim, 0, As[0], Bs[0]);

    for (int s = 0; s < nstages; ++s) {
        async_wait_all();       // my stage-s copies are in LDS
        __syncthreads();        // everyone's are; stage s-1 compute is done
        if (s + 1 < nstages) {
            stage_issue(Arows, Brows, Kdim, (s + 1) * 32,
                        As[(s + 1) & 1], Bs[(s + 1) & 1]);
        }
        float (*Ab)[36] = As[s & 1];
        float (*Bb)[36] = Bs[s & 1];

        #pragma unroll
        for (int kk = 0; kk < 32; kk += 4) {
            v2f a[2], b[2];
            #pragma unroll
            for (int mi = 0; mi < 2; ++mi) {
                int m = 32 * wm + 16 * mi + lmod;
                a[mi].x = Ab[m][kk + 2 * lhalf];
                a[mi].y = Ab[m][kk + 2 * lhalf + 1];
            }
            #pragma unroll
            for (int ni = 0; ni < 2; ++ni) {
                int n = 32 * wn + 16 * ni + lmod;
                b[ni].x = Bb[n][kk + 2 * lhalf];
                b[ni].y = Bb[n][kk + 2 * lhalf + 1];
            }
            #pragma unroll
            for (int mi = 0; mi < 2; ++mi)
                #pragma unroll
                for (int ni = 0; ni < 2; ++ni)
                    acc[mi][ni] = __builtin_amdgcn_wmma_f32_16x16x4_f32(
                        false, a[mi], false, b[ni], (short)0, acc[mi][ni],
                        false, false);
        }
    }
}

// ---------------------------------------------------------------------------
// Kernel 2 (per expert): x1 = t @ W1e^T + b1e, fused interleaved SwiGLU -> h
// x1 columns (2j, 2j+1) live in adjacent lanes of the same 16-wide tile, so
// the gate/linear pairing is a single __shfl_xor(v, 1).
// ---------------------------------------------------------------------------
__global__ void moe_gemm1_swiglu_kernel(const float* __restrict__ t,
                                        const float* __restrict__ W1e,
                                        const float* __restrict__ b1e,
                                        float* __restrict__ h) {
    __shared__ float As[2][128][36];
    __shared__ float Bs[2][64][36];
    const int tM = blockIdx.y * 128;
    const int nBase = blockIdx.x * 64;

    v8f acc[2][2];
    mma_main_loop(t + (size_t)tM * H_DIM, W1e + (size_t)nBase * H_DIM, H_DIM,
                  As, Bs, acc);

    const int lane = threadIdx.x & 31;
    const int wave = threadIdx.x >> 5;
    const int wm = wave & 3;
    const int wn = wave >> 2;
    const int lmod = lane & 15;
    const int lhalf = lane >> 4;

    #pragma unroll
    for (int mi = 0; mi < 2; ++mi) {
        #pragma unroll
        for (int ni = 0; ni < 2; ++ni) {
            const int col = nBase + 32 * wn + 16 * ni + lmod;  // x1 column
            const float bias = b1e[col];
            #pragma unroll
            for (int i = 0; i < 8; ++i) {
                const int row = tM + 32 * wm + 16 * mi + i + 8 * lhalf;
                float v = acc[mi][ni][i] + bias;
                float p = __shfl_xor(v, 1, 32);  // partner column value
                if ((lmod & 1) == 0) {
                    float xg = fminf(v, LIMIT);
                    float xl = fminf(fmaxf(p, -LIMIT), LIMIT);
                    float hv = xg * sigmoid_f(ALPHA * xg) * (xl + 1.0f);
                    h[(size_t)row * I_DIM + (col >> 1)] = hv;
                }
            }
        }
    }
}

// ---------------------------------------------------------------------------
// Kernel 3 (per expert): out += combine[:,e] * (h @ W2e^T + b2e)
// Expert kernels are serialized on the stream -> plain += is deterministic.
// ---------------------------------------------------------------------------
__global__ void moe_gemm2_combine_kernel(const float* __restrict__ h,
                                         const float* __restrict__ W2e,
                                         const float* __restrict__ b2e,
                                         const float* __restrict__ combine,
                                         int expert,
                                         float* __restrict__ out) {
    __shared__ float As[2][128][36];
    __shared__ float Bs[2][64][36];
    const int tM = blockIdx.y * 128;
    const int nBase = blockIdx.x * 64;

    v8f acc[2][2];
    mma_main_loop(h + (size_t)tM * I_DIM, W2e + (size_t)nBase * I_DIM, I_DIM,
                  As, Bs, acc);

    const int lane = threadIdx.x & 31;
    const int wave = threadIdx.x >> 5;
    const int wm = wave & 3;
    const int wn = wave >> 2;
    const int lmod = lane & 15;
    const int lhalf = lane >> 4;

    #pragma unroll
    for (int mi = 0; mi < 2; ++mi) {
        #pragma unroll
        for (int ni = 0; ni < 2; ++ni) {
            const int col = nBase + 32 * wn + 16 * ni + lmod;  // output column
            const float bias = b2e[col];
            #pragma unroll
            for (int i = 0; i < 8; ++i) {
                const int row = tM + 32 * wm + 16 * mi + i + 8 * lhalf;
                const float c = combine[(size_t)row * E_DIM + expert];
                const float v = acc[mi][ni][i] + bias;
                out[(size_t)row * H_DIM + col] += c * v;
            }
        }
    }
}

// ---------------------------------------------------------------------------
// Host-side launch
// ---------------------------------------------------------------------------
extern "C" void kernel_launch(void* const* d_in, const int* in_sizes, int n_in,
                              void* d_out, int out_size, void* d_ws, size_t ws_size,
                              hipStream_t stream) {
    (void)in_sizes; (void)n_in; (void)out_size; (void)ws_size;

    const float* x      = (const float*)d_in[0];  // [T, H]
    const float* nscale = (const float*)d_in[1];  // [H]
    const float* gate_w = (const float*)d_in[2];  // [E, H]
    const float* gate_b = (const float*)d_in[3];  // [E]
    const float* W1     = (const float*)d_in[4];  // [E, 2I, H]
    const float* b1     = (const float*)d_in[5];  // [E, 2I]
    const float* W2     = (const float*)d_in[6];  // [E, H, I]
    const float* b2     = (const float*)d_in[7];  // [E, H]
    float* out = (float*)d_out;                   // [T, H]

    // Workspace layout: t[T,H] | h[T,I] | combine[T,E]
    float* t_buf = (float*)d_ws;
    float* h_buf = t_buf + (size_t)T_DIM * H_DIM;
    float* combine = h_buf + (size_t)T_DIM * I_DIM;

    // out starts as the residual x
    (void)hipMemcpyAsync(out, x, sizeof(float) * (size_t)T_DIM * H_DIM,
                         hipMemcpyDeviceToDevice, stream);

    gate_rmsnorm_kernel<<<T_DIM, 256, 0, stream>>>(x, nscale, gate_w, gate_b,
                                                   t_buf, combine);

    const dim3 grid1(TWOI_DIM / 64, T_DIM / 128);  // 90 x 8
    const dim3 grid2(H_DIM / 64, T_DIM / 128);     // 45 x 8
    for (int e = 0; e < E_DIM; ++e) {
        const float* W1e = W1 + (size_t)e * TWOI_DIM * H_DIM;
        const float* b1e = b1 + (size_t)e * TWOI_DIM;
        const float* W2e = W2 + (size_t)e * H_DIM * I_DIM;
        const float* b2e = b2 + (size_t)e * H_DIM;
        moe_gemm1_swiglu_kernel<<<grid1, 256, 0, stream>>>(t_buf, W1e, b1e,
                                                           h_buf);
        moe_gemm2_combine_kernel<<<grid2, 256, 0, stream>>>(h_buf, W2e, b2e,
                                                            combine, e, out);
    }
}